// SpairBase_79731772883701
// MI455X (gfx1250) — compile-verified
//
#include <hip/hip_runtime.h>
#include <stdint.h>

// ----------------------------------------------------------------------------
// SPAIR object-count KL scan, MI455X (gfx1250, wave32).
//
// One wave per batch element (1024 waves).
//  - cd[0..256] in 9 VGPRs/lane (c = lane + 32*j); c>256 stays exactly 0.
//  - samples ballot'ed into 8 scalar masks; csf/denom wave-uniform scalars.
//  - ONE reduction per step (norm == p_z or 1-p_z since sum(cd)==1), done with
//    TWO chained V_WMMA_F32_16X16X4_F32 ops (exact f32):
//      wmma#1 (B=ones): D[m,n] = part[m] + part[m+16]          -> s[m]
//      8 reg adds:      lanes<16 = sum s[0..7], lanes>=16 = sum s[8..15]
//      wmma#2 (B=ones): D[m,n] = TL + TH = full 32-lane sum in ALL lanes
//  - readfirstlane(total) -> p_z uniform; KL/log/rcp chain on scalar pipe.
//  - Logs: all arguments are >= 1e-9 (eps-clamped), so the libm denormal
//    pre-scaling is dead weight; use raw v_log_f32 (log2) * ln2.
//    The input-only entropy term pi*ln(pi)+(1-pi)*ln(1-pi) is hoisted out of
//    the step loop and computed vectorized once per 32-step chunk.
// ----------------------------------------------------------------------------

#define HW 256
#define NCH 9  // 9 chunks of 32 lanes covering count support [0, 256]

typedef float v2f __attribute__((ext_vector_type(2)));
typedef float v8f __attribute__((ext_vector_type(8)));

#if defined(__has_builtin)
#if __has_builtin(__builtin_amdgcn_wmma_f32_16x16x4_f32)
#define USE_WMMA_REDUCE 1
#endif
#if __has_builtin(__builtin_amdgcn_logf)
#define HAVE_RAW_LOG 1
#endif
#endif

// ln(x) for x >= ~1e-9 (no denormal handling needed): v_log_f32 is log2.
__device__ __forceinline__ float fast_ln(float x) {
#ifdef HAVE_RAW_LOG
  return __builtin_amdgcn_logf(x) * 0.6931471805599453f;
#else
  return __logf(x);
#endif
}

__device__ __forceinline__ float wave32_sum(float x) {
#ifdef USE_WMMA_REDUCE
  v2f bm;
  bm[0] = 1.0f;
  bm[1] = 1.0f;
  // Stage 1: A vgpr0 = partials (lanes0-15:(M,K=0), lanes16-31:(M,K=2)),
  // A vgpr1 = 0 (K=1,3 rows). B = all ones => D[m,n] = part[m] + part[m+16].
  v2f a1;
  a1[0] = x;
  a1[1] = 0.0f;
  v8f c1 = {0.f, 0.f, 0.f, 0.f, 0.f, 0.f, 0.f, 0.f};
  c1 = __builtin_amdgcn_wmma_f32_16x16x4_f32(false, a1, false, bm, (short)0, c1,
                                             false, false);
  // D vgpr j: lanes<16 hold s[j], lanes>=16 hold s[j+8].
  float t = ((c1[0] + c1[1]) + (c1[2] + c1[3])) + ((c1[4] + c1[5]) + (c1[6] + c1[7]));
  // Stage 2: lanes<16 carry TL = sum s[0..7], lanes>=16 carry TH = sum s[8..15].
  // Same trick: D[m,n] = TL + TH = full sum, replicated to every lane.
  v2f a2;
  a2[0] = t;
  a2[1] = 0.0f;
  v8f c2 = {0.f, 0.f, 0.f, 0.f, 0.f, 0.f, 0.f, 0.f};
  c2 = __builtin_amdgcn_wmma_f32_16x16x4_f32(false, a2, false, bm, (short)0, c2,
                                             false, false);
  return c2[0];
#else
  for (int off = 16; off > 0; off >>= 1) x += __shfl_xor(x, off, 32);
  return x;
#endif
}

__device__ __forceinline__ float uniform_f32(float x) {
  return __int_as_float(__builtin_amdgcn_readfirstlane(__float_as_int(x)));
}

__device__ __forceinline__ float lane_f32(float v, int lane) {
  return __int_as_float(__builtin_amdgcn_readlane(__float_as_int(v), lane));
}

__global__ __launch_bounds__(256) void spair_obj_kl_scan(
    const float* __restrict__ z_pres, const float* __restrict__ z_prob,
    float* __restrict__ out) {
  const int lane = threadIdx.x & 31;
  const int wave = threadIdx.x >> 5;
  const int b = blockIdx.x * 8 + wave;

  const float* pres = z_pres + (size_t)b * HW;
  const float* prob = z_prob + (size_t)b * HW;
  float* outp = out + (size_t)b * HW;

  const float EPSL = 1e-9f;

  // gfx1250 prefetch path (global_prefetch_b8).
  __builtin_prefetch(pres + lane * 8, 0, 3);
  __builtin_prefetch(prob + lane * 8, 0, 3);

  // ---- Stage inputs: probs + per-lane entropy term, 8 scalar sample masks --
  float pr[8];   // prob for step j*32+lane
  float ent[8];  // pi*ln(pi+eps) + (1-pi)*ln(1-pi+eps), vectorized (2 logs/32 steps)
  unsigned smask[8];
#pragma unroll
  for (int j = 0; j < 8; ++j) {
    const float pi = prob[j * 32 + lane];
    pr[j] = pi;
    ent[j] = pi * fast_ln(pi + EPSL) + (1.0f - pi) * fast_ln(1.0f - pi + EPSL);
    const float zp = pres[j * 32 + lane];
    smask[j] = (unsigned)__ballot(zp >= 0.5f);  // round() (ties measure-zero)
  }

  // ---- cd0: truncated geometric, p = sigmoid(-2) ---------------------------
  // p = 1/(1+e^2); ln(p) = -2.1269280110429727; sum_{0..256}(1-p)p^c ~= 1.
  float cd[NCH], cbase[NCH];
#pragma unroll
  for (int j = 0; j < NCH; ++j) {
    const float c = (float)(lane + 32 * j);
    cbase[j] = c;
    const float v = 0.8807970779778825f * __expf(c * -2.1269280110429727f);
    cd[j] = (c <= 256.0f) ? v : 0.0f;
  }

  float fcsf = 0.0f;    // count_so_far (wave-uniform)
  float fden = 256.0f;  // HW - i      (wave-uniform)

#pragma unroll
  for (int j2 = 0; j2 < 8; ++j2) {
    const unsigned mj = smask[j2];
    const float pv = pr[j2];
    const float ev = ent[j2];
#pragma unroll 1
    for (int i2 = 0; i2 < 32; ++i2) {
      const float inv_den = __builtin_amdgcn_rcpf(fden);

      // u[c] = clamp(c - csf, 0, denom); per-lane partial of the dot cd.u
      float u[NCH];
      float part = 0.0f;
#pragma unroll
      for (int j = 0; j < NCH; ++j) {
        float t = cbase[j] - fcsf;
        t = fminf(fmaxf(t, 0.0f), fden);
        u[j] = t;
        part = fmaf(cd[j], t, part);
      }

      // Full 32-lane sum via 2x v_wmma_f32_16x16x4_f32; force uniform so the
      // KL/normalization chain scalarizes onto the scalar-float pipe.
      const float tot = uniform_f32(wave32_sum(part));
      const float p_z = tot * inv_den;  // the "bmm" result (uniform)

      // per-step scalars via readlane (uniform)
      const float pi = lane_f32(pv, i2);
      const float enti = lane_f32(ev, i2);

      // kl = ent - pi*ln(p_z+eps) - (1-pi)*ln(1-p_z+eps)
      const float lq = fast_ln(p_z + EPSL);
      const float l1q = fast_ln(1.0f - p_z + EPSL);
      const float kl = enti - pi * lq - (1.0f - pi) * l1q;
      outp[j2 * 32 + i2] = kl;  // same value/addr across lanes -> 1 transaction

      // update: norm = sample ? p_z : 1-p_z (sum(cd)==1), clamped at 1e-6
      const unsigned bit = (mj >> i2) & 1u;
      float norm = bit ? p_z : (1.0f - p_z);
      norm = fmaxf(norm, 1e-6f);
      const float scale = __builtin_amdgcn_rcpf(norm) * inv_den;

      if (bit) {  // wave-uniform scalar branch
#pragma unroll
        for (int j = 0; j < NCH; ++j) cd[j] *= u[j] * scale;
        fcsf += 1.0f;
      } else {
#pragma unroll
        for (int j = 0; j < NCH; ++j) cd[j] *= (fden - u[j]) * scale;
      }
      fden -= 1.0f;
    }
  }
}

extern "C" void kernel_launch(void* const* d_in, const int* in_sizes, int n_in,
                              void* d_out, int out_size, void* d_ws,
                              size_t ws_size, hipStream_t stream) {
  const float* z_pres = (const float*)d_in[0];  // [B,1,16,16]
  const float* z_prob = (const float*)d_in[1];  // [B,1,16,16]
  float* out = (float*)d_out;                   // [B,1,16,16]
  const int B = in_sizes[0] / HW;               // 1024
  dim3 grid(B / 8);                             // 8 waves (8 batches) / block
  dim3 block(256);
  spair_obj_kl_scan<<<grid, block, 0, stream>>>(z_pres, z_prob, out);
}